// GCNN_51848845197571
// MI455X (gfx1250) — compile-verified
//
#include <hip/hip_runtime.h>

typedef __attribute__((ext_vector_type(2))) float v2f;
typedef __attribute__((ext_vector_type(8))) float v8f;

#define FEAT 64
#define NGRAPH 64

static inline int ceildiv(long long a, long long b) { return (int)((a + b - 1) / b); }

// ---------------- degree / normalization ----------------
__global__ void k_deg_init(float* deg, int N) {
    int i = blockIdx.x * blockDim.x + threadIdx.x;
    if (i < N) deg[i] = 1.0f;            // self-loop contributes 1 to every node's degree
}

__global__ void k_deg_accum(const int* __restrict__ col, float* deg, long long E) {
    long long i = (long long)blockIdx.x * blockDim.x + threadIdx.x;
    if (i < E) atomicAdd(&deg[col[i]], 1.0f);
}

__global__ void k_deg_rsqrt(float* deg, int N) {
    int i = blockIdx.x * blockDim.x + threadIdx.x;
    if (i < N) deg[i] = rsqrtf(deg[i]);  // deg >= 1 always (self-loops)
}

// ---------------- GEMM: D[M,64] = A[M,64] @ W[64,64] via V_WMMA_F32_16X16X4_F32 ----------------
// One wave per 16x16 output tile. K-loop: 16 WMMA steps of K=4.
__global__ void k_gemm_wmma(const float* __restrict__ A, const float* __restrict__ W,
                            float* __restrict__ D, int M) {
    int gwave = (blockIdx.x * blockDim.x + threadIdx.x) >> 5;
    int lane  = threadIdx.x & 31;
    int mtiles = (M + 15) >> 4;
    int mtile = gwave >> 2;              // 4 N-tiles (64/16)
    int ntile = gwave & 3;
    if (mtile >= mtiles) return;         // uniform per-wave exit: EXEC stays all-1s for WMMA

    int m0 = mtile << 4, n0 = ntile << 4;
    int half = lane >> 4;                // 0: K=0..1 / M,N rows 0..7 ; 1: K=2..3 / rows 8..15
    int r    = lane & 15;

    int arow = m0 + r; if (arow >= M) arow = M - 1;  // clamp (M=100000 is a multiple of 16; no-op)

    v8f c = {};
#pragma unroll
    for (int k0 = 0; k0 < FEAT; k0 += 4) {
        v2f a, b;
        int ka = k0 + 2 * half;
        a.x = A[(long long)arow * FEAT + ka + 0];
        a.y = A[(long long)arow * FEAT + ka + 1];
        b.x = W[(ka + 0) * FEAT + n0 + r];
        b.y = W[(ka + 1) * FEAT + n0 + r];
        c = __builtin_amdgcn_wmma_f32_16x16x4_f32(false, a, false, b, (short)0, c, false, false);
    }
#pragma unroll
    for (int v = 0; v < 8; ++v) {
        int row = m0 + v + 8 * half;
        if (row < M) D[(long long)row * FEAT + n0 + r] = c[v];
    }
}

// ---------------- aggregation ----------------
// init accumulator with the self-loop term: out[i,f] = h[i,f] * dinv[i]^2
__global__ void k_agg_init(const float* __restrict__ h, const float* __restrict__ dinv,
                           float* __restrict__ out, long long total) {
    long long gid = (long long)blockIdx.x * blockDim.x + threadIdx.x;
    if (gid >= total) return;
    int node = (int)(gid >> 6);
    float d = dinv[node];
    out[gid] = h[gid] * d * d;
}

// 16 threads per edge, 4 features each (float4 gather, scalar f32 atomic scatter)
__global__ void k_edge_scatter(const int* __restrict__ rows, const int* __restrict__ cols,
                               const float* __restrict__ h, const float* __restrict__ dinv,
                               float* __restrict__ out, long long E) {
    long long gid  = (long long)blockIdx.x * blockDim.x + threadIdx.x;
    long long e    = gid >> 4;
    int       f4   = (int)(gid & 15);
    if (e >= E) return;
    int r = rows[e], c = cols[e];
    float nrm = dinv[r] * dinv[c];
    const float4 hv = ((const float4*)(h + (long long)r * FEAT))[f4];
    float* o = out + (long long)c * FEAT + f4 * 4;
    atomicAdd(o + 0, hv.x * nrm);
    atomicAdd(o + 1, hv.y * nrm);
    atomicAdd(o + 2, hv.z * nrm);
    atomicAdd(o + 3, hv.w * nrm);
}

__global__ void k_bias_relu(const float* __restrict__ in, const float* __restrict__ b,
                            float* __restrict__ out, long long total) {
    long long gid = (long long)blockIdx.x * blockDim.x + threadIdx.x;
    if (gid >= total) return;
    float v = in[gid] + b[gid & (FEAT - 1)];
    out[gid] = v > 0.0f ? v : 0.0f;
}

// ---------------- mean pool ----------------
__global__ void k_pool_zero(float* sums, float* counts) {
    int i = blockIdx.x * blockDim.x + threadIdx.x;
    if (i < NGRAPH * FEAT) sums[i] = 0.0f;
    if (i < NGRAPH) counts[i] = 0.0f;
}

__global__ void k_pool_accum(const float* __restrict__ h, const int* __restrict__ batch,
                             float* __restrict__ sums, float* __restrict__ counts, int N) {
    __shared__ float ls[NGRAPH * FEAT];   // 16 KB of the 320 KB WGP LDS
    __shared__ float lc[NGRAPH];
    for (int i = threadIdx.x; i < NGRAPH * FEAT; i += blockDim.x) ls[i] = 0.0f;
    for (int i = threadIdx.x; i < NGRAPH; i += blockDim.x) lc[i] = 0.0f;
    __syncthreads();

    int wave = threadIdx.x >> 5;          // 8 waves per 256-thread block
    int lane = threadIdx.x & 31;
    for (int node = blockIdx.x * 8 + wave; node < N; node += gridDim.x * 8) {
        int g = batch[node];
        atomicAdd(&ls[g * FEAT + lane],      h[(long long)node * FEAT + lane]);
        atomicAdd(&ls[g * FEAT + lane + 32], h[(long long)node * FEAT + lane + 32]);
        if (lane == 0) atomicAdd(&lc[g], 1.0f);
    }
    __syncthreads();
    for (int i = threadIdx.x; i < NGRAPH * FEAT; i += blockDim.x)
        if (ls[i] != 0.0f) atomicAdd(&sums[i], ls[i]);
    for (int i = threadIdx.x; i < NGRAPH; i += blockDim.x)
        if (lc[i] != 0.0f) atomicAdd(&counts[i], lc[i]);
}

__global__ void k_pool_div(const float* __restrict__ sums, const float* __restrict__ counts,
                           float* __restrict__ out) {
    int i = blockIdx.x * blockDim.x + threadIdx.x;
    if (i >= NGRAPH * FEAT) return;
    out[i] = sums[i] / fmaxf(counts[i >> 6], 1.0f);
}

// ---------------- driver ----------------
extern "C" void kernel_launch(void* const* d_in, const int* in_sizes, int n_in,
                              void* d_out, int out_size, void* d_ws, size_t ws_size,
                              hipStream_t stream) {
    const float* x     = (const float*)d_in[0];
    const int*   edge  = (const int*)d_in[1];
    const int*   batch = (const int*)d_in[2];
    const float* W1    = (const float*)d_in[3];
    const float* b1    = (const float*)d_in[4];
    const float* W2    = (const float*)d_in[5];
    const float* b2    = (const float*)d_in[6];
    float*       out   = (float*)d_out;

    const int       N = in_sizes[0] / FEAT;          // 100000
    const long long E = (long long)in_sizes[1] / 2;  // 3200000
    const long long NF = (long long)N * FEAT;

    // workspace layout (floats): bufA | bufB | dinv | sums | counts
    float* bufA   = (float*)d_ws;
    float* bufB   = bufA + NF;
    float* dinv   = bufB + NF;
    float* sums   = dinv + N;
    float* counts = sums + NGRAPH * FEAT;

    const int* rows = edge;       // edge_index[0]
    const int* cols = edge + E;   // edge_index[1]

    const int T = 256;

    // --- degree / D^{-1/2} (shared by both layers) ---
    k_deg_init <<<ceildiv(N, T), T, 0, stream>>>(dinv, N);
    k_deg_accum<<<ceildiv(E, T), T, 0, stream>>>(cols, dinv, E);
    k_deg_rsqrt<<<ceildiv(N, T), T, 0, stream>>>(dinv, N);

    const int mtiles   = (N + 15) / 16;
    const int gemmBlks = ceildiv((long long)mtiles * 4 * 32, T);

    // --- layer 1: bufA = x@W1 ; bufB = aggregate ; bufA = relu(bufB + b1) ---
    k_gemm_wmma   <<<gemmBlks, T, 0, stream>>>(x, W1, bufA, N);
    k_agg_init    <<<ceildiv(NF, T), T, 0, stream>>>(bufA, dinv, bufB, NF);
    k_edge_scatter<<<ceildiv(E * 16, T), T, 0, stream>>>(rows, cols, bufA, dinv, bufB, E);
    k_bias_relu   <<<ceildiv(NF, T), T, 0, stream>>>(bufB, b1, bufA, NF);

    // --- layer 2: bufB = bufA@W2 ; bufA = aggregate ; bufB = relu(bufA + b2) ---
    k_gemm_wmma   <<<gemmBlks, T, 0, stream>>>(bufA, W2, bufB, N);
    k_agg_init    <<<ceildiv(NF, T), T, 0, stream>>>(bufB, dinv, bufA, NF);
    k_edge_scatter<<<ceildiv(E * 16, T), T, 0, stream>>>(rows, cols, bufB, dinv, bufA, E);
    k_bias_relu   <<<ceildiv(NF, T), T, 0, stream>>>(bufA, b2, bufB, NF);

    // --- global mean pool ---
    k_pool_zero <<<ceildiv(NGRAPH * FEAT, T), T, 0, stream>>>(sums, counts);
    k_pool_accum<<<512, T, 0, stream>>>(bufB, batch, sums, counts, N);
    k_pool_div  <<<ceildiv(NGRAPH * FEAT, T), T, 0, stream>>>(sums, counts, out);
}